// Grad3D_7619271983642
// MI455X (gfx1250) — compile-verified
//
#include <hip/hip_runtime.h>
#include <cstdint>
#include <cstddef>

// ---------------------------------------------------------------------------
// Grad3D: out = |dx| + |dy| + |dz| of central differences with zero padding.
// Shape (N=4, C=1, D=192, H=224, W=192), f32.  Memory-bound stencil:
// ~264 MB compulsory HBM traffic -> ~11 us floor at 23.3 TB/s.
// Strategy: two TDM (tensor_load_to_lds) descriptors stage a
// (TD+2)x(TH+2)x192 halo tile into LDS (H-rows are contiguous in memory, so
// each descriptor is a clean 2-D tile), compute from LDS with b128 accesses,
// stream results out with non-temporal stores so the 132 MB input stays
// resident in the 192 MB L2.
// ---------------------------------------------------------------------------

typedef __attribute__((ext_vector_type(4))) float    fvec4;
typedef __attribute__((ext_vector_type(4))) unsigned v4u;
typedef __attribute__((ext_vector_type(8))) int      v8i;
typedef __attribute__((ext_vector_type(4))) int      v4i;

#define G3D_W  192
#define G3D_H  224
#define G3D_D  192
#define G3D_N  4
#define G3D_TD 6      // interior tile depth   (192 / 6  = 32 tiles)
#define G3D_TH 8      // interior tile height  (224 / 8  = 28 tiles)
#define G3D_NT 192    // threads per block (6 waves of 32)

#define G3D_RS   (G3D_W)                 // h-row stride (floats)
#define G3D_PS   (G3D_W * G3D_H)         // d-plane stride (43008)
#define G3D_NS   (G3D_PS * G3D_D)        // batch stride
#define G3D_LR   (G3D_TH + 2)            // LDS rows per slab (10)
#define G3D_LSLAB  (G3D_LR * G3D_W)      // floats per slab (1920)
#define G3D_LTOT   (G3D_LSLAB * (G3D_TD + 2))  // 15360 floats = 60 KB LDS
#define G3D_LROW4  (G3D_W / 4)           // 48
#define G3D_LSLAB4 (G3D_LSLAB / 4)       // 480
#define G3D_LTOT4  (G3D_LTOT / 4)        // 3840

#if defined(__HIP_DEVICE_COMPILE__) && __has_builtin(__builtin_amdgcn_tensor_load_to_lds)
#define G3D_HAS_TDM 1
#else
#define G3D_HAS_TDM 0
#endif

#if G3D_HAS_TDM
// Issue one 2-D TDM descriptor: `nslabs` slabs of LSLAB contiguous floats,
// slab-to-slab global stride = d-plane stride, packed contiguously into LDS
// starting at byte offset `lds_byte_off`.
__device__ __forceinline__ void g3d_tdm_load(const float* gptr,
                                             unsigned lds_byte_off,
                                             unsigned nslabs) {
    const uint64_t ga = (uint64_t)(uintptr_t)(const void*)gptr;
    v4u g0 = {};
    g0.x = 1u;                                           // count=1 (valid D#)
    g0.y = lds_byte_off;                                 // lds_addr (bytes)
    g0.z = (unsigned)(ga & 0xFFFFFFFFull);               // global_addr lo
    g0.w = (unsigned)((ga >> 32) & 0x01FFFFFFull)        // global_addr hi
         | (2u << 30);                                   // type = 2 ("image")

    const unsigned tdim0 = 0x40000000u;                  // huge: no OOB zeroing
    const unsigned tdim1 = 0x40000000u;
    const unsigned tile0 = (unsigned)G3D_LSLAB;          // 1920 elements
    const uint64_t str0  = (uint64_t)G3D_PS;             // dim0 stride (elems)

    v8i g1 = {};
    g1[0] = (int)(2u << 16);                             // data_size = 4 B
    g1[1] = (int)((tdim0 & 0xFFFFu) << 16);              // tensor_dim0 lo16
    g1[2] = (int)((tdim0 >> 16) | ((tdim1 & 0xFFFFu) << 16));
    g1[3] = (int)((tdim1 >> 16) | (tile0 << 16));        // tile_dim0
    g1[4] = (int)nslabs;                                 // tile_dim1, tile_dim2=0
    g1[5] = (int)(str0 & 0xFFFFFFFFull);                 // tensor_dim0_stride lo
    g1[6] = (int)((str0 >> 32) & 0xFFFFull);             // stride hi, dim1_stride=0
    g1[7] = 0;

    v4i g2 = {};
    g2[0] = 0x40000000;                                  // tensor_dim2 (unused)
    v4i g3 = {};

#if defined(__clang_major__) && (__clang_major__ >= 23)
    v8i g4 = {};
    __builtin_amdgcn_tensor_load_to_lds(g0, g1, g2, g3, g4, 0);
#else
    __builtin_amdgcn_tensor_load_to_lds(g0, g1, g2, g3, 0);
#endif
}
#endif // G3D_HAS_TDM

__global__ __launch_bounds__(G3D_NT)
void grad3d_stencil_kernel(const float* __restrict__ x, float* __restrict__ out) {
    __shared__ __align__(16) float smem[G3D_LTOT];
    fvec4* s4 = (fvec4*)smem;

    const int tid   = (int)threadIdx.x;
    const int h0    = (int)blockIdx.x * G3D_TH;
    const int d0    = (int)blockIdx.y * G3D_TD;
    const int obase = (int)blockIdx.z * G3D_NS;

#if G3D_HAS_TDM
    const bool interior =
        (d0 > 0) && (d0 + G3D_TD < G3D_D) && (h0 > 0) && (h0 + G3D_TH < G3D_H);
    if (interior) {
        // One wave issues two TDM DMAs (4 slabs each) that together cover the
        // 8-slab halo tile; the TDM engine can pipeline the two descriptors.
        if (tid < 32) {
            const float* gbase =
                x + obase + (d0 - 1) * G3D_PS + (h0 - 1) * G3D_RS;
            g3d_tdm_load(gbase, 0u, 4u);
            g3d_tdm_load(gbase + 4 * G3D_PS,
                         (unsigned)(4 * G3D_LSLAB * sizeof(float)), 4u);
        }
#if __has_builtin(__builtin_amdgcn_s_wait_tensorcnt)
        __builtin_amdgcn_s_wait_tensorcnt(0);
#else
        asm volatile("s_wait_tensorcnt 0x0" ::: "memory");
#endif
    } else
#endif // G3D_HAS_TDM
    {
        // Boundary tiles (halo crosses the zero-pad): cooperative predicated
        // b128 global->LDS fill. Also the fallback when TDM is unavailable.
        #pragma unroll 1
        for (int i = tid; i < G3D_LTOT4; i += G3D_NT) {
            const int dl  = i / G3D_LSLAB4;
            const int rem = i - dl * G3D_LSLAB4;
            const int hl  = rem / G3D_LROW4;
            const int w4  = rem - hl * G3D_LROW4;
            const int gd  = d0 - 1 + dl;
            const int gh  = h0 - 1 + hl;
            fvec4 v = {};
            if ((unsigned)gd < (unsigned)G3D_D && (unsigned)gh < (unsigned)G3D_H)
                v = *(const fvec4*)(x + obase + gd * G3D_PS + gh * G3D_RS + w4 * 4);
            s4[i] = v;
        }
    }
    __syncthreads();

    // Compute: thread owns 4 consecutive w (one fvec4), covers TD x 2 (d,h)
    // positions. 5x ds_load_b128 + 2 scalar LDS reads per 4 outputs.
    const int w4 = tid % G3D_LROW4;     // 0..47
    const int hg = tid / G3D_LROW4;     // 0..3

    #pragma unroll
    for (int hh = 0; hh < 2; ++hh) {
        const int hl = 1 + hg + 4 * hh;          // 1..8
        #pragma unroll
        for (int dl = 1; dl <= G3D_TD; ++dl) {
            const int ci = dl * G3D_LSLAB4 + hl * G3D_LROW4 + w4;
            const fvec4 c  = s4[ci];
            const fvec4 ym = s4[ci - G3D_LROW4];
            const fvec4 yp = s4[ci + G3D_LROW4];
            const fvec4 zm = s4[ci - G3D_LSLAB4];
            const fvec4 zp = s4[ci + G3D_LSLAB4];
            const int fi = ci * 4;
            const float xl = (w4 == 0)              ? 0.0f : smem[fi - 1];
            const float xr = (w4 == G3D_LROW4 - 1)  ? 0.0f : smem[fi + 4];

            fvec4 r;
            r.x = 0.5f * (fabsf(c.y - xl ) + fabsf(yp.x - ym.x) + fabsf(zp.x - zm.x));
            r.y = 0.5f * (fabsf(c.z - c.x) + fabsf(yp.y - ym.y) + fabsf(zp.y - zm.y));
            r.z = 0.5f * (fabsf(c.w - c.y) + fabsf(yp.z - ym.z) + fabsf(zp.z - zm.z));
            r.w = 0.5f * (fabsf(xr  - c.z) + fabsf(yp.w - ym.w) + fabsf(zp.w - zm.w));

            float* op = out + obase + (d0 - 1 + dl) * G3D_PS
                                    + (h0 - 1 + hl) * G3D_RS + w4 * 4;
            __builtin_nontemporal_store(r, (fvec4*)op);   // keep input hot in L2
        }
    }
}

extern "C" void kernel_launch(void* const* d_in, const int* in_sizes, int n_in,
                              void* d_out, int out_size, void* d_ws, size_t ws_size,
                              hipStream_t stream) {
    (void)in_sizes; (void)n_in; (void)out_size; (void)d_ws; (void)ws_size;
    const float* x  = (const float*)d_in[0];
    float*      out = (float*)d_out;
    dim3 grid(G3D_H / G3D_TH, G3D_D / G3D_TD, G3D_N);   // (28, 32, 4)
    dim3 block(G3D_NT);
    hipLaunchKernelGGL(grad3d_stencil_kernel, grid, block, 0, stream, x, out);
}